// OursLossFunction_45286135169500
// MI455X (gfx1250) — compile-verified
//
#include <hip/hip_runtime.h>
#include <hip/hip_bf16.h>
#include <math.h>

// ---------------------------------------------------------------------------
// OursLossFunction for MI455X (gfx1250).
// Bandwidth-bound: ~431 MB streamed once -> ~18.5us at 23.3 TB/s.
// All global traffic is staged through LDS with coalesced b128 loads; per-row
// work runs out of LDS (row stride 33 is coprime with 64 banks -> conflict
// free). Reductions are deterministic two-phase (no float atomics) and use
// V_WMMA_F32_16X16X4_F32 + ds_swizzle(SWAPX16) for exact 32-lane sums.
// ---------------------------------------------------------------------------

#define GEO_B      131072
#define S_FULL     13
#define SEQ        12          // S - 1
#define NV         33          // GEO_VOCAB + 1
#define IGNORE_IDX 32

#define BLK        256                         // mask / final block size
#define NBLK_M     ((GEO_B * S_FULL) / BLK)    // 6656, exact

#define GBLK       192                         // geo block: 192 rows = 16 batches
#define NBLK_G     ((GEO_B * SEQ) / GBLK)      // 8192, exact

// workspace layout (floats)
#define CONT_OFF   0                           // 33 floats: CONT_REWARDS
#define MP_OFF     64                          // 2 * NBLK_M mask partials
#define GP_OFF     (64 + 2 * NBLK_M)           // 3 * NBLK_G geo partials
// total: (64 + 13312 + 24576) * 4 B ~= 148 KB of d_ws

typedef __attribute__((ext_vector_type(2))) float v2f;
typedef __attribute__((ext_vector_type(8))) float v8f;

// ---------------------------------------------------------------------------
// Exact 32-lane sum on the CDNA5 matrix core.
// A (16x4 f32, 2 VGPRs): VGPR0 = x (K=0 lanes 0-15 / K=2 lanes 16-31),
// VGPR1 = 0. B = all-ones (layout independent). With C = 0:
//   D[m][n] = x_m + x_{m+16}  (replicated over n)
// Per-lane sum of the 8 D VGPRs + SWAPX16 exchange-add = total in every lane.
// Requires EXEC == all ones (all call sites are divergence-free).
// ---------------------------------------------------------------------------
__device__ __forceinline__ float wave_sum32(float x) {
#if __has_builtin(__builtin_amdgcn_wmma_f32_16x16x4_f32)
  v2f a = {x, 0.0f};
  v2f b = {1.0f, 1.0f};
  v8f c = {};
  c = __builtin_amdgcn_wmma_f32_16x16x4_f32(false, a, false, b, (short)0, c,
                                            false, false);
  float t = c[0] + c[1] + c[2] + c[3] + c[4] + c[5] + c[6] + c[7];
  int u = __builtin_amdgcn_ds_swizzle(__float_as_int(t), 0x401F); // SWAPX16
  return t + __int_as_float(u);
#else
  float t = x;
  t += __int_as_float(__builtin_amdgcn_ds_swizzle(__float_as_int(t), 0x041F));
  t += __int_as_float(__builtin_amdgcn_ds_swizzle(__float_as_int(t), 0x081F));
  t += __int_as_float(__builtin_amdgcn_ds_swizzle(__float_as_int(t), 0x101F));
  t += __int_as_float(__builtin_amdgcn_ds_swizzle(__float_as_int(t), 0x201F));
  t += __int_as_float(__builtin_amdgcn_ds_swizzle(__float_as_int(t), 0x401F));
  return t;
#endif
}

// per-row logsumexp + first-occurrence argmax out of LDS
__device__ __forceinline__ float row_lse_argmax(const float* __restrict__ L,
                                                int& am) {
  float m = L[0];
  am = 0;
#pragma unroll
  for (int j = 1; j < NV; ++j) {
    const float v = L[j];
    if (v > m) { m = v; am = j; }
  }
  float s = 0.0f;
#pragma unroll
  for (int j = 0; j < NV; ++j) s += __expf(L[j] - m);
  return m + __logf(s);
}

// ---------------------------------------------------------------------------
// Kernel 0: replicate CONT_REWARDS exactly (numpy quirk:
// x = linspace(0,b,2000) -> x_i = i*b/1999, but dx = b/2000).
// ---------------------------------------------------------------------------
__global__ void init_cont_kernel(float* ws) {
  int i = threadIdx.x;
  if (i >= NV) return;
  if (i == 0) {
    ws[CONT_OFF] = 0.0f;
  } else {
    double b = (double)i;
    double dx = b / 2000.0;
    double s = 0.0;
    for (int j = 0; j < 2000; ++j) s += pow(0.8, (double)j * b / 1999.0);
    s *= dx;
    ws[CONT_OFF + i] = (float)(1.0 / s);
  }
}

// ---------------------------------------------------------------------------
// Kernel 1: mask CE. 256 rows per block staged through LDS with coalesced
// b128 loads (block base 33792 B is 16 B aligned).
// ---------------------------------------------------------------------------
__global__ void mask_loss_kernel(const float* __restrict__ mg,
                                 const int* __restrict__ gt,
                                 float* __restrict__ ws) {
  __shared__ __align__(16) float tile[BLK * NV];   // 33792 B
  __shared__ float wred[2][BLK / 32];

  const int tid = threadIdx.x;

  // stage-in: 2112 float4 = 8448 floats, 256 threads -> 8 iters + 64-thread tail
  {
    const float4* __restrict__ src =
        (const float4*)(mg + (size_t)blockIdx.x * (BLK * NV));
    float4* __restrict__ dst = (float4*)tile;
    for (int i = tid; i < (BLK * NV) / 4; i += BLK) dst[i] = src[i];
  }
  __syncthreads();

  const float* L = tile + tid * NV;
  int am_dummy;
  const float lse = row_lse_argmax(L, am_dummy);

  const int row = blockIdx.x * BLK + tid;
  const int g = gt[row];
  const int tc = g < 0 ? 0 : (g > NV - 1 ? NV - 1 : g);
  const float valid = (g != -100) ? 1.0f : 0.0f;
  const float ce = (lse - L[tc]) * valid;

  const float ce_s = wave_sum32(ce);
  const float va_s = wave_sum32(valid);

  const int wave = tid >> 5, lane = tid & 31;
  if (lane == 0) { wred[0][wave] = ce_s; wred[1][wave] = va_s; }
  __syncthreads();
  if (tid == 0) {
    float a = 0.0f, b = 0.0f;
#pragma unroll
    for (int w = 0; w < BLK / 32; ++w) { a += wred[0][w]; b += wred[1][w]; }
    ws[MP_OFF + 2 * blockIdx.x + 0] = a;
    ws[MP_OFF + 2 * blockIdx.x + 1] = b;
  }
}

// ---------------------------------------------------------------------------
// Kernel 2: geo losses, one ROW per thread (fully coalesced). 192 threads =
// exactly 16 batch elements per block; only the per-row correct flags are
// shared. Every thread of a batch redundantly (uniformly) scans its 12 flags
// for min_idx, so there is zero divergence and EXEC stays all-ones for WMMA.
// ---------------------------------------------------------------------------
__global__ void geo_loss_kernel(const float* __restrict__ go,
                                const int* __restrict__ pos,
                                const float* __restrict__ ws_cont,
                                float* __restrict__ ws) {
  __shared__ __align__(16) float tile[GBLK * NV];  // 25344 B
  __shared__ int corr_sh[GBLK];
  __shared__ float wred[3][GBLK / 32];

  const int tid = threadIdx.x;

  // stage-in: 1584 float4 = 6336 floats (block base 25344 B, 16 B aligned)
  {
    const float4* __restrict__ src =
        (const float4*)(go + (size_t)blockIdx.x * (GBLK * NV));
    float4* __restrict__ dst = (float4*)tile;
    for (int i = tid; i < (GBLK * NV) / 4; i += GBLK) dst[i] = src[i];
  }
  __syncthreads();

  const int lb = tid / SEQ;                 // local batch 0..15
  const int t  = tid - lb * SEQ;            // timestep 0..11
  const int bg = blockIdx.x * (GBLK / SEQ) + lb;  // global batch

  const float* L = tile + tid * NV;
  int am;
  const float lse = row_lse_argmax(L, am);

  const int tgt = pos[(size_t)bg * S_FULL + t + 1];
  const int tc = tgt < 0 ? 0 : (tgt > NV - 1 ? NV - 1 : tgt);
  const float ce = (tgt == IGNORE_IDX) ? 0.0f : (lse - L[tc]);
  const int correct = (am == tgt) ? 1 : 0;

  corr_sh[tid] = correct;
  __syncthreads();

  // argmin over correct.astype(int): first wrong step, else 0 (all-ones case)
  int mi = 0;
  {
    int found = 0;
#pragma unroll
    for (int tt = 0; tt < SEQ; ++tt) {
      const int bad = 1 - corr_sh[lb * SEQ + tt];
      if (bad && !found) { mi = tt; found = 1; }
    }
  }
  const float cont = ws_cont[CONT_OFF + mi];

  // CORRECT_W[t] = 0.8^t, ERROR_W = 0.8^t + 1 ; exp2 is the native transcendental
  const float w = exp2f((float)t * -0.3219280948873623f);
  const float pfx = (t < mi) ? ce * cont : 0.0f;
  const float cor = correct ? ce * w : 0.0f;
  const float err = correct ? 0.0f : ce * (w + 1.0f);

  const float p_s = wave_sum32(pfx);
  const float c_s = wave_sum32(cor);
  const float e_s = wave_sum32(err);

  const int wave = tid >> 5, lane = tid & 31;
  if (lane == 0) { wred[0][wave] = p_s; wred[1][wave] = c_s; wred[2][wave] = e_s; }
  __syncthreads();
  if (tid == 0) {
    float a = 0.0f, bb = 0.0f, cc = 0.0f;
#pragma unroll
    for (int q = 0; q < GBLK / 32; ++q) {
      a += wred[0][q]; bb += wred[1][q]; cc += wred[2][q];
    }
    ws[GP_OFF + 3 * blockIdx.x + 0] = a;
    ws[GP_OFF + 3 * blockIdx.x + 1] = bb;
    ws[GP_OFF + 3 * blockIdx.x + 2] = cc;
  }
}

// ---------------------------------------------------------------------------
// Kernel 3: deterministic partial reduction + final 5-scalar combine.
// Trip counts (6656 = 26*256, 8192 = 32*256) are uniform -> EXEC all ones.
// ---------------------------------------------------------------------------
__global__ void final_kernel(const float* __restrict__ aux,
                             const float* __restrict__ taux,
                             const float* __restrict__ sigma,
                             const float* __restrict__ ws,
                             float* __restrict__ out) {
  __shared__ float wred[5][BLK / 32];
  const int tid = threadIdx.x;

  float s0 = 0.0f, s1 = 0.0f, s2 = 0.0f, s3 = 0.0f, s4 = 0.0f;
  for (int i = tid; i < NBLK_M; i += BLK) {
    s0 += ws[MP_OFF + 2 * i + 0];
    s1 += ws[MP_OFF + 2 * i + 1];
  }
  for (int i = tid; i < NBLK_G; i += BLK) {
    s2 += ws[GP_OFF + 3 * i + 0];
    s3 += ws[GP_OFF + 3 * i + 1];
    s4 += ws[GP_OFF + 3 * i + 2];
  }
  s0 = wave_sum32(s0); s1 = wave_sum32(s1); s2 = wave_sum32(s2);
  s3 = wave_sum32(s3); s4 = wave_sum32(s4);

  const int wave = tid >> 5, lane = tid & 31;
  if (lane == 0) {
    wred[0][wave] = s0; wred[1][wave] = s1; wred[2][wave] = s2;
    wred[3][wave] = s3; wred[4][wave] = s4;
  }
  __syncthreads();
  if (tid == 0) {
    float t[5] = {0.0f, 0.0f, 0.0f, 0.0f, 0.0f};
#pragma unroll
    for (int w = 0; w < BLK / 32; ++w)
#pragma unroll
      for (int k = 0; k < 5; ++k) t[k] += wred[k][w];

    const float mask_loss = t[0] / fmaxf(t[1], 1.0f);
    const float invN = 1.0f / ((float)GEO_B * (float)SEQ);
    const float prefix_loss  = t[2] * invN;
    const float correct_loss = t[3] * invN;
    const float error_loss   = t[4] * invN;
    const float geo_loss = prefix_loss + correct_loss + error_loss;

    const float l0 = geo_loss, l1 = mask_loss, l2 = aux[0], l3 = taux[0];
    const float g0 = sigma[0], g1 = sigma[1], g2 = sigma[2], g3 = sigma[3];
    const float weighted =
        0.5f * (l0 / (g0 * g0) + l1 / (g1 * g1) +
                l2 / (g2 * g2) + l3 / (g3 * g3)) +
        __logf(g0 * g1 * g2 * g3);

    out[0] = weighted;
    out[1] = prefix_loss;
    out[2] = correct_loss;
    out[3] = error_loss;
    out[4] = mask_loss;
  }
}

// ---------------------------------------------------------------------------
extern "C" void kernel_launch(void* const* d_in, const int* in_sizes, int n_in,
                              void* d_out, int out_size, void* d_ws,
                              size_t ws_size, hipStream_t stream) {
  (void)in_sizes; (void)n_in; (void)out_size; (void)ws_size;
  const float* geo  = (const float*)d_in[0];
  const float* mg   = (const float*)d_in[1];
  const int*   pos  = (const int*)d_in[2];
  const int*   gt   = (const int*)d_in[3];
  const float* aux  = (const float*)d_in[4];
  const float* taux = (const float*)d_in[5];
  const float* sig  = (const float*)d_in[6];
  float* out = (float*)d_out;
  float* ws  = (float*)d_ws;   // needs ~148 KB

  init_cont_kernel<<<1, 64, 0, stream>>>(ws);
  mask_loss_kernel<<<NBLK_M, BLK, 0, stream>>>(mg, gt, ws);
  geo_loss_kernel<<<NBLK_G, GBLK, 0, stream>>>(geo, pos, ws, ws);
  final_kernel<<<1, BLK, 0, stream>>>(aux, taux, sig, ws, out);
}